// RelGraphConvModel_60275571032231
// MI455X (gfx1250) — compile-verified
//
#include <hip/hip_runtime.h>
#include <hip/hip_bf16.h>
#include <math.h>

typedef __attribute__((ext_vector_type(2))) float v2f;
typedef __attribute__((ext_vector_type(8))) float v8f;

#define NN 50000
#define NE 600000
#define DIM 128
#define NREL 8
#define NBAS 8
#define NG 256
#define RD (NREL * DIM) /* 1024 */

#define BM 64 /* rows per block  (4 M-tiles) */
#define BN 64 /* cols per block  (4 N-tiles) */

// ---------------- Wcat[i][r*128+o] = sum_b comp[r,b] * bases[b][i][o] ----------------
__global__ void k_build_wcat(const float* __restrict__ bases, const float* __restrict__ comp,
                             float* __restrict__ wcat) {
  int idx = blockIdx.x * blockDim.x + threadIdx.x;
  if (idx >= DIM * RD) return;
  int i = idx / RD;
  int rem = idx - i * RD;
  int r = rem >> 7;
  int o = rem & (DIM - 1);
  float s = 0.f;
#pragma unroll
  for (int b = 0; b < NBAS; ++b)
    s += comp[r * NBAS + b] * bases[(size_t)b * DIM * DIM + (size_t)i * DIM + o];
  wcat[idx] = s;
}

// ------- Hall[N,1024] = X[N,128] @ Wcat[128,1024]  (LDS-staged f32 WMMA 16x16x4) -----
// B panel staged via async global->LDS (ASYNCcnt); A panel staged transposed for
// bank-conflict-free ds_load of A fragments. One barrier; K=128 fully resident.
__global__ void k_gemm_wmma_f32(const float* __restrict__ X, const float* __restrict__ W,
                                float* __restrict__ Y) {
  __shared__ float aT[DIM * BM]; // [k][m] transposed, 32 KB
  __shared__ float bP[DIM * BN]; // [k][n] row-major, 32 KB
  const int t = threadIdx.x;
  const int m0b = blockIdx.y * BM;
  const int n0b = blockIdx.x * BN;

  // ---- stage B via async global->LDS b128 (16B per lane-transfer) ----
  unsigned bbase = (unsigned)(size_t)(void*)bP; // low 32 bits = LDS byte offset
#pragma unroll
  for (int i = 0; i < (DIM * BN) / (4 * 256); ++i) { // 8 iters
    int lin = i * 256 + t;
    int k = lin >> 4;              // 16 float4 per 64-float row
    int c4 = (lin & 15) << 2;      // float index within row
    const float* g = W + (size_t)k * RD + n0b + c4;
    unsigned laddr = bbase + ((unsigned)(k * BN + c4) << 2);
    asm volatile("global_load_async_to_lds_b128 %0, %1, off" ::"v"(laddr), "v"(g) : "memory");
  }

  // ---- stage A transposed via regular loads + ds stores ----
#pragma unroll
  for (int i = 0; i < (BM * DIM) / (4 * 256); ++i) { // 8 iters
    int lin = i * 256 + t;   // float4 id within panel
    int m = lin >> 5;        // 32 float4 per 128-float row
    int k4 = (lin & 31) << 2;
    int row = m0b + m;
    if (row >= NN) row = NN - 1; // clamp: values unused by skipped tiles
    const float4 v = *(const float4*)(X + (size_t)row * DIM + k4);
    aT[(k4 + 0) * BM + m] = v.x;
    aT[(k4 + 1) * BM + m] = v.y;
    aT[(k4 + 2) * BM + m] = v.z;
    aT[(k4 + 3) * BM + m] = v.w;
  }
  asm volatile("s_wait_asynccnt 0" ::: "memory"); // own wave's async LDS writes done
  __syncthreads();                                // all waves' staging visible

  // ---- compute: wave w -> M-tile (w&3), N-tiles (w>>2)*2 + {0,1} ----
  const int w = t >> 5, lane = t & 31;
  const int tmL = w & 3;
  const int tnL0 = (w >> 2) * 2;
  const int tile_m0 = m0b + tmL * 16;
  if (tile_m0 + 16 <= NN) { // wave-uniform guard: EXEC all-ones for WMMA
    const int h = lane >> 4, l = lane & 15;
    v8f acc0 = {}, acc1 = {};
    const int am = tmL * 16 + l;
    const int bn0 = tnL0 * 16 + l;
    const int bn1 = bn0 + 16;
    for (int k = 0; k < DIM; k += 4) {
      int kb = k + 2 * h;
      v2f a, b0, b1;
      a.x = aT[kb * BM + am];
      a.y = aT[(kb + 1) * BM + am];
      b0.x = bP[kb * BN + bn0];
      b0.y = bP[(kb + 1) * BN + bn0];
      b1.x = bP[kb * BN + bn1];
      b1.y = bP[(kb + 1) * BN + bn1];
      acc0 = __builtin_amdgcn_wmma_f32_16x16x4_f32(false, a, false, b0, (short)0, acc0, false, false);
      acc1 = __builtin_amdgcn_wmma_f32_16x16x4_f32(false, a, false, b1, (short)0, acc1, false, false);
    }
    const int mb = tile_m0 + h * 8;
    const int n0g = n0b + tnL0 * 16 + l;
#pragma unroll
    for (int j = 0; j < 8; ++j) {
      Y[(size_t)(mb + j) * RD + n0g] = acc0[j];
      Y[(size_t)(mb + j) * RD + n0g + 16] = acc1[j];
    }
  }
}

// ---------------- agg[n,:] = bias ----------------
__global__ void k_agg_init(float* __restrict__ agg, const float* __restrict__ bias) {
  int idx = blockIdx.x * blockDim.x + threadIdx.x;
  if (idx >= NN * DIM) return;
  agg[idx] = bias[idx & (DIM - 1)];
}

// ---------------- per-edge: agg[dst] += Hall[src, etype, :] ----------------
__global__ void k_edge_scatter(const float* __restrict__ hall, const int* __restrict__ src,
                               const int* __restrict__ dst, const int* __restrict__ et,
                               float* __restrict__ agg) {
  long long tid = (long long)blockIdx.x * blockDim.x + threadIdx.x;
  if (tid >= (long long)NE * 32) return;
  int e = (int)(tid >> 5);
  int q = ((int)tid & 31) * 4;
  const float4 v = *(const float4*)(hall + (size_t)src[e] * RD + (size_t)et[e] * DIM + q);
  float* out = agg + (size_t)dst[e] * DIM + q;
  unsafeAtomicAdd(out + 0, v.x);
  unsafeAtomicAdd(out + 1, v.y);
  unsafeAtomicAdd(out + 2, v.z);
  unsafeAtomicAdd(out + 3, v.w);
}

__global__ void k_relu(float* __restrict__ x, int n) {
  int i = blockIdx.x * blockDim.x + threadIdx.x;
  if (i < n) x[i] = fmaxf(x[i], 0.f);
}

// ---------------- attention pooling ----------------
__device__ __forceinline__ unsigned enc_f(float f) {
  unsigned b = __float_as_uint(f);
  return (b & 0x80000000u) ? ~b : (b | 0x80000000u);
}
__device__ __forceinline__ float dec_f(unsigned k) {
  unsigned b = (k & 0x80000000u) ? (k ^ 0x80000000u) : ~k;
  return __uint_as_float(b);
}

__global__ void k_pool_init(unsigned* __restrict__ segm_enc, float* __restrict__ segs,
                            float* __restrict__ readout) {
  int i = blockIdx.x * blockDim.x + threadIdx.x;
  if (i < NG * DIM) readout[i] = 0.f;
  if (i < NG) {
    segm_enc[i] = 0x007FFFFFu; // enc(-inf)
    segs[i] = 0.f;
  }
}

// one wave per node: gate = h . gate_w + gate_b ; atomic segment max
__global__ void k_gate(const float* __restrict__ h, const float* __restrict__ gw,
                       const float* __restrict__ gb, const int* __restrict__ n2g,
                       float* __restrict__ gate, unsigned* __restrict__ segm_enc) {
  int n = (blockIdx.x * blockDim.x + threadIdx.x) >> 5;
  int lane = threadIdx.x & 31;
  if (n >= NN) return;
  const float* row = h + (size_t)n * DIM;
  float s = 0.f;
#pragma unroll
  for (int j = 0; j < 4; ++j) s += row[lane + 32 * j] * gw[lane + 32 * j];
#pragma unroll
  for (int o = 16; o > 0; o >>= 1) s += __shfl_xor(s, o, 32);
  if (lane == 0) {
    float g = s + gb[0];
    gate[n] = g;
    atomicMax(&segm_enc[n2g[n]], enc_f(g));
  }
}

__global__ void k_expsum(const float* __restrict__ gate, const int* __restrict__ n2g,
                         const unsigned* __restrict__ segm_enc, float* __restrict__ expg,
                         float* __restrict__ segs) {
  int n = blockIdx.x * blockDim.x + threadIdx.x;
  if (n >= NN) return;
  int g = n2g[n];
  float e = expf(gate[n] - dec_f(segm_enc[g]));
  expg[n] = e;
  unsafeAtomicAdd(&segs[g], e);
}

__global__ void k_readout(const float* __restrict__ h, const float* __restrict__ expg,
                          const float* __restrict__ segs, const int* __restrict__ n2g,
                          float* __restrict__ readout) {
  long long tid = (long long)blockIdx.x * blockDim.x + threadIdx.x;
  if (tid >= (long long)NN * 32) return;
  int n = (int)(tid >> 5);
  int q = ((int)tid & 31) * 4;
  int g = n2g[n];
  float wgt = expg[n] / segs[g];
  const float4 v = *(const float4*)(h + (size_t)n * DIM + q);
  float* out = readout + (size_t)g * DIM + q;
  unsafeAtomicAdd(out + 0, v.x * wgt);
  unsafeAtomicAdd(out + 1, v.y * wgt);
  unsafeAtomicAdd(out + 2, v.z * wgt);
  unsafeAtomicAdd(out + 3, v.w * wgt);
}

// ---------------- MLP head: one block(128) per graph ----------------
__global__ void k_mlp(const float* __restrict__ readout, const float* __restrict__ fc1w,
                      const float* __restrict__ fc1b, const float* __restrict__ fc2w,
                      const float* __restrict__ fc2b, const float* __restrict__ fc3w,
                      const float* __restrict__ fc3b, float* __restrict__ out) {
  __shared__ float ro[DIM];
  __shared__ float z1[100];
  __shared__ float z2[64];
  int g = blockIdx.x;
  int t = threadIdx.x;
  ro[t] = readout[(size_t)g * DIM + t];
  __syncthreads();
  if (t < 100) {
    float s = fc1b[t];
    for (int i = 0; i < DIM; ++i) s += ro[i] * fc1w[i * 100 + t];
    z1[t] = fmaxf(s, 0.f);
  }
  __syncthreads();
  if (t < 64) {
    float s = fc2b[t];
    for (int i = 0; i < 100; ++i) s += z1[i] * fc2w[i * 64 + t];
    z2[t] = fmaxf(s, 0.f);
  }
  __syncthreads();
  if (t == 0) {
    float s = fc3b[0];
    for (int i = 0; i < 64; ++i) s += z2[i] * fc3w[i];
    out[g] = 1.f / (1.f + expf(-s));
  }
}

extern "C" void kernel_launch(void* const* d_in, const int* in_sizes, int n_in,
                              void* d_out, int out_size, void* d_ws, size_t ws_size,
                              hipStream_t stream) {
  (void)in_sizes; (void)n_in; (void)out_size; (void)ws_size;
  const float* features = (const float*)d_in[0];
  const int* src = (const int*)d_in[1];
  const int* dst = (const int*)d_in[2];
  const int* etype = (const int*)d_in[3];
  const int* n2g = (const int*)d_in[4];
  const float* bases[3] = {(const float*)d_in[5], (const float*)d_in[8], (const float*)d_in[11]};
  const float* comp[3] = {(const float*)d_in[6], (const float*)d_in[9], (const float*)d_in[12]};
  const float* bias[3] = {(const float*)d_in[7], (const float*)d_in[10], (const float*)d_in[13]};
  const float* gw = (const float*)d_in[14];
  const float* gb = (const float*)d_in[15];
  const float* fc1w = (const float*)d_in[16];
  const float* fc1b = (const float*)d_in[17];
  const float* fc2w = (const float*)d_in[18];
  const float* fc2b = (const float*)d_in[19];
  const float* fc3w = (const float*)d_in[20];
  const float* fc3b = (const float*)d_in[21];
  float* out = (float*)d_out;

  char* ws = (char*)d_ws;
  size_t off = 0;
  auto take = [&](size_t bytes) {
    char* p = ws + off;
    off = (off + bytes + 255) & ~(size_t)255;
    return p;
  };
  float* wcat = (float*)take((size_t)DIM * RD * 4);
  float* hall = (float*)take((size_t)NN * RD * 4);
  float* bufA = (float*)take((size_t)NN * DIM * 4);
  float* bufB = (float*)take((size_t)NN * DIM * 4);
  float* gate = (float*)take((size_t)NN * 4);
  float* expg = (float*)take((size_t)NN * 4);
  unsigned* segm_enc = (unsigned*)take((size_t)NG * 4);
  float* segs = (float*)take((size_t)NG * 4);
  float* readout = (float*)take((size_t)NG * DIM * 4);

  const int B = 256;
  const dim3 gemm_grid(RD / BN, (NN / 16 + (BM / 16) - 1) / (BM / 16)); // 16 x 782
  float* bufs[2] = {bufA, bufB};
  const float* X = features;
  for (int L = 0; L < 3; ++L) {
    k_build_wcat<<<(DIM * RD + B - 1) / B, B, 0, stream>>>(bases[L], comp[L], wcat);
    k_gemm_wmma_f32<<<gemm_grid, B, 0, stream>>>(X, wcat, hall);
    float* agg = bufs[L & 1];
    k_agg_init<<<(NN * DIM + B - 1) / B, B, 0, stream>>>(agg, bias[L]);
    k_edge_scatter<<<(int)(((long long)NE * 32 + B - 1) / B), B, 0, stream>>>(hall, src, dst,
                                                                              etype, agg);
    if (L < 2) k_relu<<<(NN * DIM + B - 1) / B, B, 0, stream>>>(agg, NN * DIM);
    X = agg;
  }
  const float* h3 = bufs[0]; // layer 3 wrote bufA

  k_pool_init<<<(NG * DIM + B - 1) / B, B, 0, stream>>>(segm_enc, segs, readout);
  k_gate<<<(NN * 32 + B - 1) / B, B, 0, stream>>>(h3, gw, gb, n2g, gate, segm_enc);
  k_expsum<<<(NN + B - 1) / B, B, 0, stream>>>(gate, n2g, segm_enc, expg, segs);
  k_readout<<<(int)(((long long)NN * 32 + B - 1) / B), B, 0, stream>>>(h3, expg, segs, n2g,
                                                                       readout);
  k_mlp<<<NG, DIM, 0, stream>>>(readout, fc1w, fc1b, fc2w, fc2b, fc3w, fc3b, out);
}